// GCN_51651276702106
// MI455X (gfx1250) — compile-verified
//
#include <hip/hip_runtime.h>

typedef float v2f __attribute__((ext_vector_type(2)));
typedef float v4f __attribute__((ext_vector_type(4)));
typedef float v8f __attribute__((ext_vector_type(8)));

#define FEAT 128

// ---------------- degree / dinv ----------------

__global__ void k_init_deg(float* deg, int n) {
    int i = blockIdx.x * blockDim.x + threadIdx.x;
    if (i < n) deg[i] = 1.0f;  // self-loop contributes 1
}

__global__ void k_deg(const long long* __restrict__ dst, float* __restrict__ deg, int E) {
    int e = blockIdx.x * blockDim.x + threadIdx.x;
    if (e < E) unsafeAtomicAdd(&deg[(int)dst[e]], 1.0f);
}

__global__ void k_dinv(float* deg, int n) {
    int i = blockIdx.x * blockDim.x + threadIdx.x;
    if (i < n) {
        float d = deg[i];
        deg[i] = (d > 0.0f) ? rsqrtf(d) : 0.0f;
    }
}

__global__ void k_zero(float* __restrict__ p, long n) {
    long i = (long)blockIdx.x * blockDim.x + threadIdx.x;
    if (i < n) p[i] = 0.0f;
}

// ---------------- dense GEMM: H = act(X) * W^T + b  (WMMA f32 16x16x4) ----------------
// One wave computes one 16x16 output tile; 8 waves/block cover all 128 output cols
// for one 16-row stripe. K loop: 32 x V_WMMA_F32_16X16X4_F32.

template <bool RELU_IN>
__global__ void k_gemm(const float* __restrict__ X, const float* __restrict__ W,
                       const float* __restrict__ bias, float* __restrict__ H, int M) {
    const int m_tile = blockIdx.x;
    const int wave   = threadIdx.x >> 5;
    const int lane   = threadIdx.x & 31;
    const int n_base = wave * 16;

    const int mr = lane & 15;   // row within A tile / col within B tile
    const int kh = lane >> 4;   // which K pair this half-wave holds

    int row = m_tile * 16 + mr;
    if (row >= M) row = M - 1;  // clamp loads; stores guarded below

    const float* __restrict__ xrow = X + (size_t)row * FEAT;
    const float* __restrict__ wrow = W + (size_t)(n_base + mr) * FEAT;

    v8f c = {};
#pragma unroll
    for (int k0 = 0; k0 < FEAT; k0 += 4) {
        v2f a = *(const v2f*)(xrow + k0 + 2 * kh);
        if (RELU_IN) { a.x = fmaxf(a.x, 0.0f); a.y = fmaxf(a.y, 0.0f); }
        v2f b = *(const v2f*)(wrow + k0 + 2 * kh);
        // 8 args: (neg_a, A, neg_b, B, c_mod, C, reuse_a, reuse_b)
        c = __builtin_amdgcn_wmma_f32_16x16x4_f32(false, a, false, b, (short)0, c,
                                                  false, false);
    }

    const int col = n_base + mr;
    const float bv = bias[col];
#pragma unroll
    for (int i = 0; i < 8; ++i) {
        int orow = m_tile * 16 + i + 8 * kh;
        if (orow < M) H[(size_t)orow * FEAT + col] = c[i] + bv;
    }
}

// ---------------- edge scatter: agg[dst] += h[src] * dinv[src]*dinv[dst] ----------------
// One wave per edge; lane handles 4 contiguous features (128 = 32 lanes * 4).
// Edges [0,E) from edge list; edges [E, E+N) are self loops.

__global__ void k_scatter(const float* __restrict__ H, const long long* __restrict__ src,
                          const long long* __restrict__ dst, const float* __restrict__ dinv,
                          float* __restrict__ agg, int E, int N) {
    const long long e = (long long)blockIdx.x * (blockDim.x >> 5) + (threadIdx.x >> 5);
    const int lane = threadIdx.x & 31;
    if (e >= (long long)E + N) return;

    int s, d;
    if (e < E) {
        s = (int)src[e];
        d = (int)dst[e];
    } else {
        s = d = (int)(e - E);
    }
    const float norm = dinv[s] * dinv[d];

    const float* __restrict__ hs = H + (size_t)s * FEAT + lane * 4;
    float* __restrict__ ad = agg + (size_t)d * FEAT + lane * 4;

    v4f v = *(const v4f*)hs;  // global_load_b128, L2-resident
    unsafeAtomicAdd(ad + 0, v.x * norm);
    unsafeAtomicAdd(ad + 1, v.y * norm);
    unsafeAtomicAdd(ad + 2, v.z * norm);
    unsafeAtomicAdd(ad + 3, v.w * norm);
}

// ---------------- launch ----------------

extern "C" void kernel_launch(void* const* d_in, const int* in_sizes, int n_in,
                              void* d_out, int out_size, void* d_ws, size_t ws_size,
                              hipStream_t stream) {
    const float*     x   = (const float*)d_in[0];
    const long long* ei  = (const long long*)d_in[1];  // int64 [2, E]
    const float*     W1  = (const float*)d_in[2];
    const float*     b1  = (const float*)d_in[3];
    const float*     W2  = (const float*)d_in[4];
    const float*     b2  = (const float*)d_in[5];
    float*           out = (float*)d_out;

    const int N = in_sizes[0] / FEAT;  // 50000
    const int E = in_sizes[1] / 2;     // 1,600,000
    const long long* srcp = ei;
    const long long* dstp = ei + E;

    float* ws   = (float*)d_ws;
    float* dinv = ws;                        // N floats (deg, then dinv in place)
    float* h    = ws + 65536;                // N*FEAT floats
    float* agg  = h + (size_t)N * FEAT;      // N*FEAT floats

    const long nf = (long)N * FEAT;
    const int  m_tiles = (N + 15) / 16;
    const int  total_e = E + N;
    const int  scat_blocks = (total_e + 7) / 8;  // 8 waves (edges) per 256-thread block

    // degree -> dinv (shared by both layers)
    k_init_deg<<<(N + 255) / 256, 256, 0, stream>>>(dinv, N);
    k_deg<<<(E + 255) / 256, 256, 0, stream>>>(dstp, dinv, E);
    k_dinv<<<(N + 255) / 256, 256, 0, stream>>>(dinv, N);

    // layer 1: h = x W1^T + b1 ; agg = A_hat h
    k_gemm<false><<<m_tiles, 256, 0, stream>>>(x, W1, b1, h, N);
    k_zero<<<(int)((nf + 255) / 256), 256, 0, stream>>>(agg, nf);
    k_scatter<<<scat_blocks, 256, 0, stream>>>(h, srcp, dstp, dinv, agg, E, N);

    // layer 2: h = relu(agg) W2^T + b2 ; out = A_hat h
    k_gemm<true><<<m_tiles, 256, 0, stream>>>(agg, W2, b2, h, N);
    k_zero<<<(int)((nf + 255) / 256), 256, 0, stream>>>(out, nf);
    k_scatter<<<scat_blocks, 256, 0, stream>>>(h, srcp, dstp, dinv, out, E, N);
}